// SimplePoseGCN_3393024164007
// MI455X (gfx1250) — compile-verified
//
#include <hip/hip_runtime.h>
#include <hip/hip_bf16.h>

typedef __attribute__((ext_vector_type(2))) float v2f;
typedef __attribute__((ext_vector_type(8))) float v8f;

// Fused SimplePoseGCN forward for gfx1250 (wave32, WMMA).
// Grid: B/G workgroups of 256 threads (8 waves). Each WG owns G=8 batch
// elements; activations live in LDS; the three 128x128 GEMMs run on
// V_WMMA_F32_16X16X4_F32 (exact fp32).
__global__ void posegcn_fused_kernel(
    const float* __restrict__ x,    // [B,17,2]
    const float* __restrict__ adj,  // [17,17]
    const float* __restrict__ w1, const float* __restrict__ b1,  // [2,128],[128]
    const float* __restrict__ w2, const float* __restrict__ b2,  // [128,128],[128]
    const float* __restrict__ w3, const float* __restrict__ b3,
    const float* __restrict__ w4, const float* __restrict__ b4,
    const float* __restrict__ w5, const float* __restrict__ b5,  // [128,3],[3]
    float* __restrict__ out)        // [B,17,3]
{
    constexpr int N    = 17;
    constexpr int H    = 128;
    constexpr int G    = 8;          // batch elements per workgroup
    constexpr int ROWS = G * N;      // 136 real activation rows
    constexpr int RPAD = 144;        // padded to 9 tiles of 16
    constexpr int AS   = 130;        // LDS row stride (dwords): breaks bank conflicts

    extern __shared__ float smem[];
    float* act  = smem;                      // [RPAD][AS]
    float* ytmp = smem + RPAD * AS;          // [RPAD][AS]
    float* adjs = smem + 2 * RPAD * AS;      // [17*17] (+pad)
    float* xs   = adjs + 292;                // [ROWS*2]

    const int tid  = threadIdx.x;
    const int lane = tid & 31;
    const int wave = tid >> 5;               // 0..7  -> owns N-strip n0..n0+15
    const int col  = lane & 15;
    const int half = lane >> 4;              // lane 0-15 vs 16-31
    const int n0   = wave * 16;
    const int g0   = blockIdx.x * G;

    // ---- stage adj and x into LDS ----
    for (int e = tid; e < N * N; e += 256)    adjs[e] = adj[e];
    for (int e = tid; e < ROWS * 2; e += 256) xs[e]   = x[(size_t)g0 * N * 2 + e];
    __syncthreads();

    // ---- layer 1: act = x @ w1 + b1  (K=2, pure VALU) ----
    {
        const int d = tid & 127;             // feature fixed per thread (step 256)
        const float w10 = w1[d], w11 = w1[H + d], bb = b1[d];
        for (int e = tid; e < ROWS * H; e += 256) {
            const int row = e >> 7;
            act[row * AS + d] = fmaf(xs[row * 2], w10,
                                fmaf(xs[row * 2 + 1], w11, bb));
        }
    }
    __syncthreads();

    // ---- three GCN blocks ----
    const float* Ws[3] = { w2, w3, w4 };
    const float* Bs[3] = { b2, b3, b4 };
    for (int layer = 0; layer < 3; ++layer) {
        const float* __restrict__ W  = Ws[layer];
        const float* __restrict__ Bv = Bs[layer];

        // Preload this wave's B panel (128 x 16 of W) into 64 VGPRs.
        // B 4x16 layout: VGPR0 = rows {k0, k0+2}, VGPR1 = rows {k0+1, k0+3},
        // lanes 0-15 / 16-31 hold the two K halves, column = n0+col.
        v2f breg[32];
#pragma unroll
        for (int kk = 0; kk < 32; ++kk) {
            const int kb = 4 * kk + 2 * half;
            v2f t;
            t.x = W[kb * H + n0 + col];
            t.y = W[(kb + 1) * H + n0 + col];
            breg[kk] = t;
        }
        const float bias = Bv[n0 + col];

        // GEMM: ytmp = act @ W + bias, 16x16 tiles, K=128 in 32 WMMA steps.
        for (int mt = 0; mt < RPAD / 16; ++mt) {
            const int m0 = mt * 16;
            v8f c;
#pragma unroll
            for (int r = 0; r < 8; ++r) c[r] = bias;   // same column for all 8 acc VGPRs

            // A 16x4 layout: lane row = m0+col, lanes 0-15 take K{k0,k0+1},
            // lanes 16-31 take K{k0+2,k0+3}: one ds_load_b64 per lane per step.
            const float* ab = &act[(m0 + col) * AS + 2 * half];
#pragma unroll
            for (int kk = 0; kk < 32; ++kk) {
                v2f a = *(const v2f*)(ab + 4 * kk);
                c = __builtin_amdgcn_wmma_f32_16x16x4_f32(
                        false, a, false, breg[kk], (short)0, c, false, false);
            }
            // C layout: VGPR r -> row m0 + r + 8*half, column n0+col.
#pragma unroll
            for (int r = 0; r < 8; ++r)
                ytmp[(m0 + 8 * half + r) * AS + n0 + col] = c[r];
        }
        __syncthreads();

        // adj-mix + residual + ReLU (in place on act).
        // Each thread owns (batch g, feature d) columns: load the 17 y rows
        // once, emit all 17 mixed rows (adj is an LDS scalar broadcast).
        for (int cc = tid; cc < G * H; cc += 256) {
            const int g = cc >> 7;
            const int d = cc & 127;
            float yv[N];
#pragma unroll
            for (int j = 0; j < N; ++j) yv[j] = ytmp[(g * N + j) * AS + d];
#pragma unroll
            for (int i = 0; i < N; ++i) {
                float acc = act[(g * N + i) * AS + d];   // residual
#pragma unroll
                for (int j = 0; j < N; ++j) acc = fmaf(adjs[i * N + j], yv[j], acc);
                act[(g * N + i) * AS + d] = fmaxf(acc, 0.0f);
            }
        }
        __syncthreads();
    }

    // ---- head: out = act @ w5 + b5  (128 -> 3) ----
    for (int e = tid; e < ROWS * 3; e += 256) {
        const int row = e / 3;
        const int c3  = e % 3;
        float acc = b5[c3];
#pragma unroll 8
        for (int k = 0; k < H; ++k)
            acc = fmaf(act[row * AS + k], w5[k * 3 + c3], acc);
        out[(size_t)g0 * N * 3 + e] = acc;
    }
}

extern "C" void kernel_launch(void* const* d_in, const int* in_sizes, int n_in,
                              void* d_out, int out_size, void* d_ws, size_t ws_size,
                              hipStream_t stream) {
    (void)in_sizes; (void)n_in; (void)d_ws; (void)ws_size; (void)out_size;
    const float* x   = (const float*)d_in[0];
    const float* adj = (const float*)d_in[1];
    const float* w1  = (const float*)d_in[2];
    const float* b1  = (const float*)d_in[3];
    const float* w2  = (const float*)d_in[4];
    const float* b2  = (const float*)d_in[5];
    const float* w3  = (const float*)d_in[6];
    const float* b3  = (const float*)d_in[7];
    const float* w4  = (const float*)d_in[8];
    const float* b4  = (const float*)d_in[9];
    const float* w5  = (const float*)d_in[10];
    const float* b5  = (const float*)d_in[11];
    float* out = (float*)d_out;

    constexpr int B = 32768, G = 8;
    constexpr int RPAD = 144, AS = 130;
    const size_t shmem = (size_t)(2 * RPAD * AS + 292 + 8 * 17 * 2) * sizeof(float); // ~152 KB

    dim3 grid(B / G), block(256);
    posegcn_fused_kernel<<<grid, block, shmem, stream>>>(
        x, adj, w1, b1, w2, b2, w3, b3, w4, b4, w5, b5, out);
}